// Attn_70798240907336
// MI455X (gfx1250) — compile-verified
//
#include <hip/hip_runtime.h>
#include <math.h>

typedef __attribute__((ext_vector_type(2))) float v2f;
typedef __attribute__((ext_vector_type(8))) float v8f;

#define HAVE_WMMA_F32_4 __has_builtin(__builtin_amdgcn_wmma_f32_16x16x4_f32)

static constexpr int kB = 64;     // batch
static constexpr int kS = 2048;   // sequence
static constexpr int kD = 1024;   // feature dim
static constexpr int kTileS = 16;
static constexpr int kTiles = kS / kTileS;  // 128 s-tiles per batch

// ---------------------------------------------------------------------------
// Kernel 1: energies[b, s] = dot(h[b, :], E[s, b, :])
// One wave32 per (b, 16-row s-tile). WMMA f32 16x16x4:
//   A (16x4)  = h[b, d0..d0+3] broadcast to all 16 rows (documented layout:
//               slot (vgpr j, lane-half h) holds k = 2h + j)
//   B (4x16)  = E[s0+n, b, d0..d0+3]  (same slot->k convention)
//   C (16x16) accumulates; every row/column equals energy(s0+n).
// ---------------------------------------------------------------------------
__global__ __launch_bounds__(256) void attn_energies_kernel(
    const float* __restrict__ H,        // [kB, kD]
    const float* __restrict__ E,        // [kS, kB, kD]
    float* __restrict__ energies)       // [kB, kS]
{
  const int wave = (blockIdx.x * blockDim.x + threadIdx.x) >> 5;  // 0..8191
  const int lane = threadIdx.x & 31;
  const int b    = wave >> 7;                 // wave / kTiles
  const int s0   = (wave & (kTiles - 1)) << 4;
  const int n    = lane & 15;                 // column (s within tile)
  const int half = lane >> 4;                 // lane-half -> k base 2*half

  const float* __restrict__ erow = E + ((size_t)(s0 + n) * kB + b) * kD + half * 2;
  const float* __restrict__ hrow = H + (size_t)b * kD + half * 2;

#if HAVE_WMMA_F32_4
  v8f c = {};
#else
  float p = 0.0f;
#endif

  for (int d0 = 0; d0 < kD; d0 += 16) {
    // stay ~1KB ahead on this row (speculative; dropped if unmapped)
    __builtin_prefetch(erow + d0 + 256, 0, 0);
#pragma unroll
    for (int q = 0; q < 16; q += 4) {
      v2f eb = *(const v2f*)(erow + d0 + q);   // B slots: E[d0+q+2h], E[d0+q+2h+1]
      v2f hb = *(const v2f*)(hrow + d0 + q);   // A slots: h[d0+q+2h], h[d0+q+2h+1]
#if HAVE_WMMA_F32_4
      // 8 args: (neg_a, A, neg_b, B, c_mod, C, reuse_a, reuse_b)
      c = __builtin_amdgcn_wmma_f32_16x16x4_f32(
          false, hb, false, eb, (short)0, c, false, false);
#else
      p = fmaf(eb.x, hb.x, fmaf(eb.y, hb.y, p));
#endif
    }
  }

#if HAVE_WMMA_F32_4
  const float val = c[0];                  // D[0, n] (all rows identical)
#else
  const float val = p + __shfl_xor(p, 16); // combine k-halves
#endif
  if (lane < 16)
    energies[(size_t)b * kS + s0 + lane] = val;
}

// ---------------------------------------------------------------------------
// Kernel 2: softmax over s per batch. 64 blocks x 256 threads, 8 elems/thread.
// ---------------------------------------------------------------------------
__global__ __launch_bounds__(256) void attn_softmax_kernel(
    const float* __restrict__ energies,  // [kB, kS]
    float* __restrict__ out)             // [kB, 1, kS]
{
  const int b = blockIdx.x;
  const int t = threadIdx.x;
  const int lane = t & 31;
  const int wid  = t >> 5;
  const float* __restrict__ row = energies + (size_t)b * kS + (size_t)t * 8;

  float4 e0 = *(const float4*)(row);
  float4 e1 = *(const float4*)(row + 4);

  __shared__ float red[8];

  // ---- block max ----
  float m = fmaxf(fmaxf(fmaxf(e0.x, e0.y), fmaxf(e0.z, e0.w)),
                  fmaxf(fmaxf(e1.x, e1.y), fmaxf(e1.z, e1.w)));
  for (int o = 16; o; o >>= 1) m = fmaxf(m, __shfl_xor(m, o));
  if (lane == 0) red[wid] = m;
  __syncthreads();
  if (t < 8) {
    float v = red[t];
    for (int o = 4; o; o >>= 1) v = fmaxf(v, __shfl_xor(v, o));
    if (t == 0) red[0] = v;
  }
  __syncthreads();
  m = red[0];
  __syncthreads();  // red is reused below

  // ---- exp + block sum ----
  e0.x = expf(e0.x - m); e0.y = expf(e0.y - m);
  e0.z = expf(e0.z - m); e0.w = expf(e0.w - m);
  e1.x = expf(e1.x - m); e1.y = expf(e1.y - m);
  e1.z = expf(e1.z - m); e1.w = expf(e1.w - m);
  float s = (e0.x + e0.y + e0.z + e0.w) + (e1.x + e1.y + e1.z + e1.w);
  for (int o = 16; o; o >>= 1) s += __shfl_xor(s, o);
  if (lane == 0) red[wid] = s;
  __syncthreads();
  if (t < 8) {
    float v = red[t];
    for (int o = 4; o; o >>= 1) v += __shfl_xor(v, o);
    if (t == 0) red[0] = v;
  }
  __syncthreads();
  const float inv = 1.0f / red[0];

  e0.x *= inv; e0.y *= inv; e0.z *= inv; e0.w *= inv;
  e1.x *= inv; e1.y *= inv; e1.z *= inv; e1.w *= inv;

  float* __restrict__ o_ = out + (size_t)b * kS + (size_t)t * 8;
  *(float4*)(o_)     = e0;
  *(float4*)(o_ + 4) = e1;
}

// ---------------------------------------------------------------------------
extern "C" void kernel_launch(void* const* d_in, const int* in_sizes, int n_in,
                              void* d_out, int out_size, void* d_ws, size_t ws_size,
                              hipStream_t stream) {
  (void)in_sizes; (void)n_in; (void)out_size; (void)ws_size;
  const float* H = (const float*)d_in[0];  // hidden [1, 64, 1024]
  const float* E = (const float*)d_in[1];  // encoder_outputs [2048, 64, 1024]
  float* out      = (float*)d_out;         // [64, 1, 2048]
  float* energies = (float*)d_ws;          // needs 64*2048*4 = 512 KB

  const int total_waves = kB * kTiles;                 // 8192
  const int grid1 = (total_waves * 32) / 256;          // 1024 blocks
  attn_energies_kernel<<<grid1, 256, 0, stream>>>(H, E, energies);
  attn_softmax_kernel<<<kB, 256, 0, stream>>>(energies, out);
}